// GAT_origin_87479893885513
// MI455X (gfx1250) — compile-verified
//
#include <hip/hip_runtime.h>

// ---------------- problem constants (match reference) ----------------
#define B_    32
#define N_    512
#define FIN   16
#define H_    8
#define DH    8
#define DOUT  32
#define ALPHA 0.2f

typedef __attribute__((ext_vector_type(2))) float v2f;
typedef __attribute__((ext_vector_type(8))) float v8f;

// ---------------- workspace layout (floats) ----------------
#define OFF_WH    0                              // [B,H,N,DH]   1,048,576
#define OFF_F1    (OFF_WH   + B_*H_*N_*DH)       // [B,H,N]        131,072
#define OFF_F2    (OFF_F1   + B_*H_*N_)          // [B,H,N]        131,072
#define OFF_HCAT  (OFF_F2   + B_*H_*N_)          // [B,N,64]     1,048,576
#define OFF_WHO   (OFF_HCAT + B_*N_*H_*DH)       // [B,N,32]       524,288
#define OFF_G1    (OFF_WHO  + B_*N_*DOUT)        // [B,N]           16,384
#define OFF_G2    (OFF_G1   + B_*N_)             // [B,N]           16,384
#define OFF_OUT2  (OFF_G2   + B_*N_)             // [B,N,32]       524,288
#define WS_FLOATS (OFF_OUT2 + B_*N_*DOUT)        // total ~3.44M floats (~13.8 MB)

#define WMMA_F32(A, Bv, C) \
  __builtin_amdgcn_wmma_f32_16x16x4_f32(false, (A), false, (Bv), (short)0, (C), false, false)

// ======================================================================
// K1: Wh[b,h,n,d] = xv[b,n,:] @ W_heads[h,:,d];  f1/f2 = Wh . a1/a2
// grid (H, B), block 128
// ======================================================================
__global__ void gat_k1_wh(const float* __restrict__ xv,
                          const float* __restrict__ W_heads,
                          const float* __restrict__ a1,
                          const float* __restrict__ a2,
                          float* __restrict__ ws) {
  const int h = blockIdx.x, b = blockIdx.y;
  __shared__ float Wsh[FIN][DH];
  __shared__ float a1s[DH], a2s[DH];
  const int t = threadIdx.x;
  if (t < FIN * DH) Wsh[t / DH][t % DH] = W_heads[h * FIN * DH + t];
  if (t < DH) { a1s[t] = a1[h * DH + t]; a2s[t] = a2[h * DH + t]; }
  __syncthreads();

  float* Wh = ws + OFF_WH + (size_t)(b * H_ + h) * N_ * DH;
  float* f1 = ws + OFF_F1 + (size_t)(b * H_ + h) * N_;
  float* f2 = ws + OFF_F2 + (size_t)(b * H_ + h) * N_;

  for (int n = t; n < N_; n += blockDim.x) {
    float x[FIN];
    const float* xr = xv + ((size_t)b * N_ + n) * FIN;
#pragma unroll
    for (int f = 0; f < FIN; ++f) x[f] = xr[f];
    float s1 = 0.f, s2 = 0.f;
#pragma unroll
    for (int d = 0; d < DH; ++d) {
      float acc = 0.f;
#pragma unroll
      for (int f = 0; f < FIN; ++f) acc += x[f] * Wsh[f][d];
      Wh[n * DH + d] = acc;
      s1 += acc * a1s[d];
      s2 += acc * a2s[d];
    }
    f1[n] = s1;
    f2[n] = s2;
  }
}

// ======================================================================
// K2: fused GAT layer 1 for one (b, h, 16-row tile).
// p_ij = adj>0 ? exp(leakyrelu(f1_i+f2_j)) : 0 ; h_i = elu( (p@Wh)_i / sum_j p_ij )
// One wave per block; numerator via V_WMMA_F32_16X16X4_F32.
// B fragments for a whole 64-wide j-block are staged in registers first so
// their load clause overlaps the 16-chunk softmax/WMMA chain (no per-chunk
// load->wait->wmma stall).  C cols 8..15 are dont-care (never stored).
// grid (N/16, H, B), block 32
// ======================================================================
__global__ void gat_k2_layer1(const float* __restrict__ adj,
                              const float* __restrict__ wsc,
                              float* __restrict__ ws) {
  const int ib = blockIdx.x, h = blockIdx.y, b = blockIdx.z;
  const int lane  = threadIdx.x;
  const int row16 = lane & 15;   // A-row / B,C-column
  const int khalf = lane >> 4;   // which k-half of the 4-chunk
  const int i0 = ib * 16;
  const int i  = i0 + row16;

  const float* Wh = wsc + OFF_WH + (size_t)(b * H_ + h) * N_ * DH;
  const float* f1 = wsc + OFF_F1 + (size_t)(b * H_ + h) * N_;
  const float* f2 = wsc + OFF_F2 + (size_t)(b * H_ + h) * N_;
  const float* adjRow = adj + ((size_t)b * N_ + i) * N_;

  const float f1v = f1[i];
  const int   d   = row16 & 7;     // always in-bounds; cols 8..15 are dont-care

  // per-lane base pointers: inner-loop offsets become immediates
  const float* adjK = adjRow + khalf * 2;
  const float* f2K  = f2 + khalf * 2;
  const float* WhK  = Wh + (khalf * 2) * DH + d;

  v8f   c   = {};
  float den = 0.f;

  for (int jb = 0; jb < N_; jb += 64) {
    __builtin_prefetch(adjK + jb + 64, 0, 0);
    // stage all 16 B fragments of this block (one big load clause)
    v2f bfr[16];
#pragma unroll
    for (int u = 0; u < 16; ++u) {
      bfr[u].x = WhK[(jb + 4 * u) * DH];
      bfr[u].y = WhK[(jb + 4 * u) * DH + DH];
    }
#pragma unroll
    for (int u = 0; u < 16; ++u) {
      const int j0 = jb + 4 * u;
      const v2f am = *(const v2f*)(adjK + j0);
      const v2f fj = *(const v2f*)(f2K + j0);
      float s0 = f1v + fj.x; s0 = s0 > 0.f ? s0 : ALPHA * s0;
      float s1 = f1v + fj.y; s1 = s1 > 0.f ? s1 : ALPHA * s1;
      const float p0 = am.x > 0.f ? __expf(s0) : 0.f;
      const float p1 = am.y > 0.f ? __expf(s1) : 0.f;
      den += p0 + p1;
      v2f A;  A.x = p0; A.y = p1;               // A[row16, jA..jA+1]
      c = WMMA_F32(A, bfr[u], c);
    }
  }

  // row denominator: combine the two k-halves covering the same row
  den += __shfl_xor(den, 16);

  float* hcat = ws + OFF_HCAT + (size_t)b * N_ * (H_ * DH);
#pragma unroll
  for (int r = 0; r < 8; ++r) {
    const float dlo = __shfl(den, r);
    const float dhi = __shfl(den, r + 8);
    float dm = khalf ? dhi : dlo;
    dm = dm > 0.f ? dm : 1.f;                   // isolated-node guard
    const int M = r + 8 * khalf;                // C row held in c[r]
    float v = c[r] / dm;
    v = v > 0.f ? v : (__expf(v) - 1.f);        // elu
    if (row16 < 8) hcat[(size_t)(i0 + M) * (H_ * DH) + h * DH + d] = v;
  }
}

// ======================================================================
// K3: Who = h_cat @ W_out ; g1/g2 = Who . a1_out/a2_out.  One row per thread.
// ======================================================================
__global__ void gat_k3_who(const float* __restrict__ W_out,
                           const float* __restrict__ a1o,
                           const float* __restrict__ a2o,
                           float* __restrict__ ws) {
  const int tid = blockIdx.x * blockDim.x + threadIdx.x;
  if (tid >= B_ * N_) return;
  const float* hrow = ws + OFF_HCAT + (size_t)tid * (H_ * DH);
  float hl[H_ * DH];
#pragma unroll
  for (int f = 0; f < H_ * DH; ++f) hl[f] = hrow[f];
  float* who = ws + OFF_WHO + (size_t)tid * DOUT;
  float g1 = 0.f, g2 = 0.f;
#pragma unroll 4
  for (int dd = 0; dd < DOUT; ++dd) {
    float acc = 0.f;
#pragma unroll
    for (int f = 0; f < H_ * DH; ++f) acc += hl[f] * W_out[f * DOUT + dd];
    who[dd] = acc;
    g1 += acc * a1o[dd];
    g2 += acc * a2o[dd];
  }
  ws[OFF_G1 + tid] = g1;
  ws[OFF_G2 + tid] = g2;
}

// ======================================================================
// K4: fused GAT layer 2 (D_out=32 -> two 16-col WMMA tiles).
// 32-wide j-blocks keep the staged B fragments at 16 v2f regs.
// grid (N/16, B), block 32
// ======================================================================
__global__ void gat_k4_layer2(const float* __restrict__ adj,
                              const float* __restrict__ wsc,
                              float* __restrict__ ws) {
  const int ib = blockIdx.x, b = blockIdx.y;
  const int lane  = threadIdx.x;
  const int row16 = lane & 15;
  const int khalf = lane >> 4;
  const int i0 = ib * 16;
  const int i  = i0 + row16;

  const float* Who = wsc + OFF_WHO + (size_t)b * N_ * DOUT;
  const float* g1  = wsc + OFF_G1  + (size_t)b * N_;
  const float* g2  = wsc + OFF_G2  + (size_t)b * N_;
  const float* adjRow = adj + ((size_t)b * N_ + i) * N_;
  const float g1v = g1[i];

  const float* adjK = adjRow + khalf * 2;
  const float* g2K  = g2 + khalf * 2;
  const float* WhoK = Who + (khalf * 2) * DOUT + row16;   // +0 / +16 for the two tiles

  v8f c0 = {}, c1 = {};
  float den = 0.f;

  for (int jb = 0; jb < N_; jb += 32) {
    __builtin_prefetch(adjK + jb + 64, 0, 0);
    v2f b0[8], b1[8];
#pragma unroll
    for (int u = 0; u < 8; ++u) {
      const int j0 = jb + 4 * u;
      b0[u].x = WhoK[j0 * DOUT];        b0[u].y = WhoK[j0 * DOUT + DOUT];
      b1[u].x = WhoK[j0 * DOUT + 16];   b1[u].y = WhoK[j0 * DOUT + DOUT + 16];
    }
#pragma unroll
    for (int u = 0; u < 8; ++u) {
      const int j0 = jb + 4 * u;
      const v2f am = *(const v2f*)(adjK + j0);
      const v2f gj = *(const v2f*)(g2K + j0);
      float s0 = g1v + gj.x; s0 = s0 > 0.f ? s0 : ALPHA * s0;
      float s1 = g1v + gj.y; s1 = s1 > 0.f ? s1 : ALPHA * s1;
      const float p0 = am.x > 0.f ? __expf(s0) : 0.f;
      const float p1 = am.y > 0.f ? __expf(s1) : 0.f;
      den += p0 + p1;
      v2f A;  A.x = p0; A.y = p1;
      c0 = WMMA_F32(A, b0[u], c0);
      c1 = WMMA_F32(A, b1[u], c1);
    }
  }

  den += __shfl_xor(den, 16);

  float* out2 = ws + OFF_OUT2 + (size_t)b * N_ * DOUT;
#pragma unroll
  for (int r = 0; r < 8; ++r) {
    const float dlo = __shfl(den, r);
    const float dhi = __shfl(den, r + 8);
    float dm = khalf ? dhi : dlo;
    dm = dm > 0.f ? dm : 1.f;
    const int M = r + 8 * khalf;
    float va = c0[r] / dm; va = va > 0.f ? va : (__expf(va) - 1.f);
    float vb = c1[r] / dm; vb = vb > 0.f ? vb : (__expf(vb) - 1.f);
    out2[(size_t)(i0 + M) * DOUT +      row16] = va;
    out2[(size_t)(i0 + M) * DOUT + 16 + row16] = vb;
  }
}

// ======================================================================
// K5: q = out.reshape(B, N*DOUT) @ W_q + b_q   (M=32, N=512, K=16384), WMMA.
// grid (N/16, 2), block 32
// ======================================================================
__global__ void gat_k5_q(const float* __restrict__ W_q,
                         const float* __restrict__ b_q,
                         const float* __restrict__ wsc,
                         float* __restrict__ qout) {
  const int nt = blockIdx.x, mt = blockIdx.y;
  const int lane  = threadIdx.x;
  const int row16 = lane & 15;
  const int khalf = lane >> 4;
  const int n0 = nt * 16, m0 = mt * 16;

  const float* arow = wsc + OFF_OUT2 + (size_t)(m0 + row16) * (N_ * DOUT) + khalf * 2;
  const float* bcol = W_q + (size_t)(khalf * 2) * N_ + (n0 + row16);

  v8f c = {};
  for (int kb = 0; kb < N_ * DOUT; kb += 64) {
    __builtin_prefetch(arow + kb + 64, 0, 0);
    // stage the 16 strided W_q fragments of this block first
    v2f bfr[16];
#pragma unroll
    for (int u = 0; u < 16; ++u) {
      const size_t k0 = (size_t)(kb + 4 * u);
      bfr[u].x = bcol[k0 * N_];
      bfr[u].y = bcol[k0 * N_ + N_];
    }
#pragma unroll
    for (int u = 0; u < 16; ++u) {
      const v2f A = *(const v2f*)(arow + kb + 4 * u); // out_flat[b, kA..kA+1]
      c = WMMA_F32(A, bfr[u], c);
    }
  }

  const float bq = b_q[n0 + row16];
#pragma unroll
  for (int r = 0; r < 8; ++r) {
    const int M = r + 8 * khalf;
    qout[(size_t)(m0 + M) * N_ + n0 + row16] = c[r] + bq;
  }
}

// ======================================================================
extern "C" void kernel_launch(void* const* d_in, const int* in_sizes, int n_in,
                              void* d_out, int out_size, void* d_ws, size_t ws_size,
                              hipStream_t stream) {
  const float* xv      = (const float*)d_in[0];  // [B,N,FIN]
  const float* adj     = (const float*)d_in[1];  // [B,N,N]
  const float* W_heads = (const float*)d_in[2];  // [H,FIN,DH]
  const float* a1      = (const float*)d_in[3];  // [H,DH]
  const float* a2      = (const float*)d_in[4];  // [H,DH]
  const float* W_out   = (const float*)d_in[5];  // [H*DH, DOUT]
  const float* a1_out  = (const float*)d_in[6];  // [DOUT]
  const float* a2_out  = (const float*)d_in[7];  // [DOUT]
  const float* W_q     = (const float*)d_in[8];  // [N*DOUT, N]
  const float* b_q     = (const float*)d_in[9];  // [N]
  float* out = (float*)d_out;                    // [B,N,1]
  float* ws  = (float*)d_ws;                     // needs WS_FLOATS*4 bytes (~13.8MB)
  (void)in_sizes; (void)n_in; (void)out_size; (void)ws_size;

  gat_k1_wh    <<<dim3(H_, B_),       128, 0, stream>>>(xv, W_heads, a1, a2, ws);
  gat_k2_layer1<<<dim3(N_/16, H_, B_), 32, 0, stream>>>(adj, ws, ws);
  gat_k3_who   <<<dim3((B_*N_+255)/256), 256, 0, stream>>>(W_out, a1_out, a2_out, ws);
  gat_k4_layer2<<<dim3(N_/16, B_),     32, 0, stream>>>(adj, ws, ws);
  gat_k5_q     <<<dim3(N_/16, 2),      32, 0, stream>>>(W_q, b_q, ws, out);
}